// LinearCriterion_43147241455920
// MI455X (gfx1250) — compile-verified
//
#include <hip/hip_runtime.h>
#include <math.h>

#define B_   256
#define N_   65536
#define E_   128
#define K_   8
#define LPAD 132                    // LDS row stride (floats): bank-conflict-free B frags
#define INV_T 14.285714285714286f   // 1/0.07 ; also the fixed softmax shift (|dot|<=1)
#define HPW  0.1f

typedef __attribute__((ext_vector_type(2))) float v2f;
typedef __attribute__((ext_vector_type(8))) float v8f;

// workspace layout (float offsets)
#define WS_M    0      // [256] row max of logits
#define WS_S    256    // [256] row sumexp of logits
#define WS_S2   512    // [256] sum_n exp(hp_logits - 1/T)
#define WS_KLD  768    // [256] per-b kld partial
#define WS_W    1024   // [256] is_aff flag as float
#define WS_ACC  1280   // [0]=sum inst loss, [1]=sum aff loss
#define WS_FEA  1536   // [256*128] fea_hp

__global__ void k_init(float* __restrict__ ws) {
  int i = blockIdx.x * blockDim.x + threadIdx.x;
  if (i < 1024) ws[512 + i] = 0.0f;   // zero s2, kld, w, acc
}

// ---------------- Kernel 1: per-row online softmax stats over logits ----------------
__global__ void __launch_bounds__(256) k_rowstats(const float* __restrict__ logits,
                                                  float* __restrict__ ws) {
  int b = blockIdx.x, t = threadIdx.x;
  const float* row = logits + (size_t)b * N_;
  float m = -3.4e38f, s = 0.0f;
  for (int n = t; n < N_; n += 256) {
    float x = row[n];
    if (x > m) { s *= __expf(m - x); m = x; }
    s += __expf(x - m);
  }
  __shared__ float sm[256], ss[256];
  sm[t] = m; ss[t] = s;
  __syncthreads();
  for (int off = 128; off > 0; off >>= 1) {
    if (t < off) {
      float m1 = sm[t], s1 = ss[t], m2 = sm[t + off], s2 = ss[t + off];
      float mm = fmaxf(m1, m2);
      ss[t] = s1 * __expf(m1 - mm) + s2 * __expf(m2 - mm);
      sm[t] = mm;
    }
    __syncthreads();
  }
  if (t == 0) { ws[WS_M + b] = sm[0]; ws[WS_S + b] = ss[0]; }
}

// ---------------- Kernel 2: hard-positive selection + inst/aff losses ----------------
__global__ void k_hpsel(const float* __restrict__ logits, const float* __restrict__ memory,
                        const int* __restrict__ index, const int* __restrict__ aff_idx,
                        const int* __restrict__ aff_counts, float* __restrict__ ws) {
  int b = blockIdx.x;
  int lane = threadIdx.x;                  // 32 threads = one wave
  int y = index[b];
  int cnt = aff_counts[y];
  float fi[4];
#pragma unroll
  for (int j = 0; j < 4; j++) fi[j] = memory[(size_t)y * E_ + lane + 32 * j];
  int nb[K_];
#pragma unroll
  for (int k = 0; k < K_; k++) nb[k] = aff_idx[(size_t)y * K_ + k];
  float sims[K_];
#pragma unroll
  for (int k = 0; k < K_; k++) {
    float s = 0.0f;
#pragma unroll
    for (int j = 0; j < 4; j++) s += memory[(size_t)nb[k] * E_ + lane + 32 * j] * fi[j];
#pragma unroll
    for (int off = 16; off > 0; off >>= 1) s += __shfl_down(s, off, 32);
    sims[k] = __shfl(s, 0, 32);
  }
  int hp = -1; float best = -3.4e38f;
#pragma unroll
  for (int k = 0; k < K_; k++) {
    bool valid = (k < cnt) && (nb[k] != y);
    if (valid && sims[k] > best) { best = sims[k]; hp = k; }   // strict > : first argmax
  }
  bool isaff = (cnt > 1);
  int hpn = (isaff && hp >= 0) ? nb[hp] : -1;
  float* fh = ws + WS_FEA + (size_t)b * E_;
#pragma unroll
  for (int j = 0; j < 4; j++)
    fh[lane + 32 * j] = (hpn >= 0) ? memory[(size_t)hpn * E_ + lane + 32 * j] : 0.0f;
  if (lane == 0) {
    ws[WS_W + b] = isaff ? 1.0f : 0.0f;
    float m = ws[WS_M + b], s = ws[WS_S + b];
    if (isaff) {
      float sp = 0.0f;
      for (int k = 0; k < K_; k++)
        if (k < cnt) sp += __expf(logits[(size_t)b * N_ + nb[k]] - m);
      atomicAdd(&ws[WS_ACC + 1], -(__logf(sp) - __logf(s)));   // -log(sum_p/s)
    } else {
      float pl = logits[(size_t)b * N_ + y] - m - __logf(s);   // log p_self
      atomicAdd(&ws[WS_ACC + 0], -pl);
    }
  }
}

// ---------------- Kernel 3: WMMA pass A — s2[b] = sum_n exp(dot/T - 1/T) ----------------
__global__ void __launch_bounds__(256) k_hp_sumexp(const float* __restrict__ memory,
                                                   float* __restrict__ ws) {
  extern __shared__ float lds[];
  const float* fea = ws + WS_FEA;
  int tid = threadIdx.x;
  for (int i = tid; i < B_ * E_; i += 256)
    lds[(i >> 7) * LPAD + (i & 127)] = fea[i];
  __syncthreads();
  int lane = tid & 31, wv = tid >> 5;
  int lr = lane & 15, half = lane >> 4;
  float sloc[16];
#pragma unroll
  for (int i = 0; i < 16; i++) sloc[i] = 0.0f;

  int gw = blockIdx.x * 8 + wv;
  int nwaves = gridDim.x * 8;
  for (int nt = gw; nt < N_ / 16; nt += nwaves) {
    int n0 = nt * 16;
    // A fragments: lane owns memory row n0+lr; half-select folded into the address:
    // lane needs exactly cols {4k+2*half, 4k+2*half+1}  ->  one global_load_b64 each.
    const float2* mrow2 =
        reinterpret_cast<const float2*>(memory + (size_t)(n0 + lr) * E_ + half * 2);
    v2f afr[32];
#pragma unroll
    for (int k = 0; k < 32; k++) {
      float2 a = mrow2[2 * k];
      v2f t; t.x = a.x; t.y = a.y;
      afr[k] = t;
    }
    int ntn = nt + nwaves;                 // prefetch next tile's A rows
    if (ntn < N_ / 16)
      __builtin_prefetch(memory + (size_t)(ntn * 16 + lr) * E_ + half * 2, 0, 3);

    for (int bt = 0; bt < 16; ++bt) {
      v8f acc = {};
      const float2* bp =
          reinterpret_cast<const float2*>(lds + (bt * 16 + lr) * LPAD + half * 2);
#pragma unroll
      for (int k = 0; k < 32; k++) {
        float2 bv = bp[2 * k];
        v2f bf; bf.x = bv.x; bf.y = bv.y;
        acc = __builtin_amdgcn_wmma_f32_16x16x4_f32(false, afr[k], false, bf,
                                                    (short)0, acc, false, false);
      }
      float cs = 0.0f;
#pragma unroll
      for (int r = 0; r < 8; r++) cs += __expf(acc[r] * INV_T - INV_T);
      cs += __shfl_down(cs, 16, 32);       // fold rows 8..15 (upper half-wave) into lower
      if (half == 0) sloc[bt] += cs;
    }
  }
  if (half == 0) {
#pragma unroll
    for (int bt = 0; bt < 16; ++bt) atomicAdd(&ws[WS_S2 + bt * 16 + lr], sloc[bt]);
  }
}

// ---------------- Kernel 4: WMMA pass B — fused KLD row-sum ----------------
// term = exp(L - m_b)/s_b * (L + delta_b - dot/T), delta_b = -m_b - log s_b + 1/T + log s2_b
__global__ void __launch_bounds__(256) k_hp_kld(const float* __restrict__ memory,
                                                const float* __restrict__ logits,
                                                float* __restrict__ ws) {
  extern __shared__ float lds[];
  const float* fea = ws + WS_FEA;
  int tid = threadIdx.x;
  for (int i = tid; i < B_ * E_; i += 256)
    lds[(i >> 7) * LPAD + (i & 127)] = fea[i];
  __syncthreads();
  int lane = tid & 31, wv = tid >> 5;
  int lr = lane & 15, half = lane >> 4;

  float mbA[16], isbA[16], dltA[16], kloc[16];
#pragma unroll
  for (int bt = 0; bt < 16; ++bt) {
    int colb = bt * 16 + lr;
    float mb = ws[WS_M + colb];
    float sb = ws[WS_S + colb];
    float s2b = ws[WS_S2 + colb];
    mbA[bt] = mb;
    isbA[bt] = 1.0f / sb;
    dltA[bt] = -mb - __logf(sb) + INV_T + __logf(s2b);
    kloc[bt] = 0.0f;
  }

  int gw = blockIdx.x * 8 + wv;
  int nwaves = gridDim.x * 8;
  for (int nt = gw; nt < N_ / 16; nt += nwaves) {
    int n0 = nt * 16;
    const float2* mrow2 =
        reinterpret_cast<const float2*>(memory + (size_t)(n0 + lr) * E_ + half * 2);
    v2f afr[32];
#pragma unroll
    for (int k = 0; k < 32; k++) {
      float2 a = mrow2[2 * k];
      v2f t; t.x = a.x; t.y = a.y;
      afr[k] = t;
    }
    int ntn = nt + nwaves;                 // prefetch next tile's A rows
    if (ntn < N_ / 16)
      __builtin_prefetch(memory + (size_t)(ntn * 16 + lr) * E_ + half * 2, 0, 3);

    for (int bt = 0; bt < 16; ++bt) {
      v8f acc = {};
      const float2* bp =
          reinterpret_cast<const float2*>(lds + (bt * 16 + lr) * LPAD + half * 2);
#pragma unroll
      for (int k = 0; k < 32; k++) {
        float2 bv = bp[2 * k];
        v2f bf; bf.x = bv.x; bf.y = bv.y;
        acc = __builtin_amdgcn_wmma_f32_16x16x4_f32(false, afr[k], false, bf,
                                                    (short)0, acc, false, false);
      }
      int colb = bt * 16 + lr;
      const float* lp = logits + (size_t)colb * N_ + n0 + half * 8;  // C rows: M=r+8*half
      float mb = mbA[bt], isb = isbA[bt], dlt = dltA[bt];
      float cs = 0.0f;
#pragma unroll
      for (int r = 0; r < 8; r++) {
        float L = lp[r];
        cs += __expf(L - mb) * isb * (L + dlt - acc[r] * INV_T);
      }
      cs += __shfl_down(cs, 16, 32);
      if (half == 0) kloc[bt] += cs;
    }
  }
  if (half == 0) {
#pragma unroll
    for (int bt = 0; bt < 16; ++bt) atomicAdd(&ws[WS_KLD + bt * 16 + lr], kloc[bt]);
  }
}

// ---------------- Kernel 5: finalize the 4 scalars ----------------
__global__ void k_final(const float* __restrict__ ws, float* __restrict__ out) {
  __shared__ float red[256];
  int t = threadIdx.x;
  red[t] = ws[WS_W + t] * ws[WS_KLD + t];
  __syncthreads();
  for (int off = 128; off > 0; off >>= 1) {
    if (t < off) red[t] += red[t + off];
    __syncthreads();
  }
  if (t == 0) {
    float l_hp   = HPW * red[0] / (float)B_;
    float l_inst = ws[WS_ACC + 0] / (float)B_;
    float l_aff  = ws[WS_ACC + 1] / (float)B_;
    out[0] = l_inst + l_aff + l_hp;
    out[1] = l_inst;
    out[2] = l_aff;
    out[3] = l_hp;
  }
}

extern "C" void kernel_launch(void* const* d_in, const int* in_sizes, int n_in,
                              void* d_out, int out_size, void* d_ws, size_t ws_size,
                              hipStream_t stream) {
  const float* logits     = (const float*)d_in[0];
  const float* memory     = (const float*)d_in[1];
  const int*   index      = (const int*)d_in[2];
  const int*   aff_idx    = (const int*)d_in[3];
  const int*   aff_counts = (const int*)d_in[4];
  float* ws  = (float*)d_ws;
  float* out = (float*)d_out;

  k_init<<<4, 256, 0, stream>>>(ws);
  k_rowstats<<<B_, 256, 0, stream>>>(logits, ws);
  k_hpsel<<<B_, 32, 0, stream>>>(logits, memory, index, aff_idx, aff_counts, ws);

  size_t shm = (size_t)LPAD * B_ * sizeof(float);   // 135168 B of LDS (fea_hp, padded)
  k_hp_sumexp<<<128, 256, shm, stream>>>(memory, ws);
  k_hp_kld<<<128, 256, shm, stream>>>(memory, logits, ws);

  k_final<<<1, 256, 0, stream>>>(ws, out);
}